// FinancialGNN_47519518163663
// MI455X (gfx1250) — compile-verified
//
#include <hip/hip_runtime.h>
#include <hip/hip_bf16.h>
#include <cstddef>

typedef float v2f __attribute__((ext_vector_type(2)));
typedef float v8f __attribute__((ext_vector_type(8)));

#define NEG_SLOPE 0.2f
#define BN_INVS   0.9999950000374997f   /* 1/sqrt(1+1e-5) */

// ---------- helpers ----------
__device__ __forceinline__ unsigned ord_f32(float x) {
  unsigned u = __float_as_uint(x);
  return (u & 0x80000000u) ? ~u : (u | 0x80000000u);
}
__device__ __forceinline__ float deord_f32(unsigned u) {
  return __uint_as_float((u & 0x80000000u) ? (u & 0x7fffffffu) : ~u);
}
__device__ __forceinline__ float wave_sum(float v) {
  for (int m = 16; m > 0; m >>= 1) v += __shfl_xor(v, m, 32);
  return v;
}

// ---------- fills ----------
__global__ void fill_f32(float* __restrict__ p, float v, long n) {
  long i = (long)blockIdx.x * blockDim.x + threadIdx.x;
  if (i < n) p[i] = v;
}
__global__ void fill_u32(unsigned* __restrict__ p, unsigned v, long n) {
  long i = (long)blockIdx.x * blockDim.x + threadIdx.x;
  if (i < n) p[i] = v;
}

// ---------- fp32 WMMA GEMM: C[MxNO] = act(A[MxK] @ W[KxNO] + bias) ----------
// One wave computes a 64x32 supertile (4 M-tiles x 2 N-tiles) with
// V_WMMA_F32_16X16X4_F32. K and NO are compile-time so all fragment loads
// fold into base+immediate-offset addressing; the K-loop is software-
// pipelined (prefetch k+4 while the 8 WMMAs of step k issue).
// act: 0 = none, 1 = relu
template <int K, int NO>
__global__ void gemm_f32_wmma(const float* __restrict__ A, const float* __restrict__ W,
                              const float* __restrict__ bias, float* __restrict__ C,
                              int M, int act) {
  int wave = blockIdx.x * (blockDim.x >> 5) + (threadIdx.x >> 5);
  int lane = threadIdx.x & 31;
  int mtiles = M >> 4;
  int smt = (mtiles + 3) >> 2;     // M supertiles (64 rows each, last may be ragged)
  constexpr int snt = NO >> 5;     // N supertiles (32 cols each)
  if (wave >= smt * snt) return;
  int sm = wave / snt;
  int sn = wave % snt;

  int r    = lane & 15;            // row (A) / col (B,C) within 16-wide tile
  int hi   = lane >> 4;            // half selector
  int koff = hi << 1;              // lanes 16-31 hold K = k+2, k+3

  const float* arow[4];
  bool mvalid[4];
#pragma unroll
  for (int j = 0; j < 4; ++j) {
    int mt = sm * 4 + j;
    mvalid[j] = (mt < mtiles);
    int mtc = mvalid[j] ? mt : (mtiles - 1);     // clamp ragged tail (EXEC stays full)
    arow[j] = A + (size_t)(mtc * 16 + r) * K + koff;
  }
  int n0 = sn * 32;
  const float* wb = W + (size_t)koff * NO + n0 + r;   // step-k fragment at +k*NO

  v8f acc[4][2] = {};
  v2f a[4], b0, b1;

  // prologue: fragments for k = 0
  b0.x = wb[0];       b0.y = wb[NO];
  b1.x = wb[16];      b1.y = wb[NO + 16];
#pragma unroll
  for (int j = 0; j < 4; ++j) { a[j].x = arow[j][0]; a[j].y = arow[j][1]; }

#pragma unroll
  for (int k = 0; k < K; k += 4) {
    v2f an[4], bn0, bn1;
    if (k + 4 < K) {           // prefetch next step's fragments
      const float* wn = wb + (size_t)(k + 4) * NO;
      bn0.x = wn[0];   bn0.y = wn[NO];
      bn1.x = wn[16];  bn1.y = wn[NO + 16];
#pragma unroll
      for (int j = 0; j < 4; ++j) { an[j].x = arow[j][k + 4]; an[j].y = arow[j][k + 5]; }
    }
#pragma unroll
    for (int j = 0; j < 4; ++j) {
      acc[j][0] = __builtin_amdgcn_wmma_f32_16x16x4_f32(false, a[j], false, b0,
                                                        (short)0, acc[j][0], false, false);
      acc[j][1] = __builtin_amdgcn_wmma_f32_16x16x4_f32(false, a[j], false, b1,
                                                        (short)0, acc[j][1], false, false);
    }
    if (k + 4 < K) {
      b0 = bn0; b1 = bn1;
#pragma unroll
      for (int j = 0; j < 4; ++j) a[j] = an[j];
    }
  }

  int col0 = n0 + r;
  int col1 = n0 + 16 + r;
  float bv0 = bias ? bias[col0] : 0.0f;
  float bv1 = bias ? bias[col1] : 0.0f;
#pragma unroll
  for (int j = 0; j < 4; ++j) {
    if (!mvalid[j]) continue;
    int row0 = (sm * 4 + j) * 16 + (hi << 3);
#pragma unroll
    for (int v = 0; v < 8; ++v) {
      float x0 = acc[j][0][v] + bv0;
      float x1 = acc[j][1][v] + bv1;
      if (act == 1) { x0 = fmaxf(x0, 0.0f); x1 = fmaxf(x1, 0.0f); }
      C[(size_t)(row0 + v) * NO + col0] = x0;
      C[(size_t)(row0 + v) * NO + col1] = x1;
    }
  }
}

// ---------- edge-attr mean reduction ----------
__global__ void ea_reduce(const float* __restrict__ ea, float* __restrict__ easum, int E) {
  __shared__ float s0[256], s1[256], s2[256];
  int t = threadIdx.x;
  float a = 0.f, b = 0.f, c = 0.f;
  for (int e = blockIdx.x * 256 + t; e < E; e += gridDim.x * 256) {
    a += ea[(size_t)e * 3 + 0];
    b += ea[(size_t)e * 3 + 1];
    c += ea[(size_t)e * 3 + 2];
  }
  s0[t] = a; s1[t] = b; s2[t] = c;
  __syncthreads();
  for (int st = 128; st > 0; st >>= 1) {
    if (t < st) { s0[t] += s0[t + st]; s1[t] += s1[t + st]; s2[t] += s2[t + st]; }
    __syncthreads();
  }
  if (t == 0) {
    atomicAdd(&easum[0], s0[0]);
    atomicAdd(&easum[1], s1[0]);
    atomicAdd(&easum[2], s2[0]);
  }
}

// ---------- per-layer edge-attention projection: watt = W_edge @ att_edge ----------
__global__ void prep_watt(const float* __restrict__ easum, float invE,
                          const float* We0, const float* ae0,
                          const float* We1, const float* ae1,
                          const float* We2, const float* ae2,
                          float* __restrict__ wattOut /* 12 floats */) {
  if (threadIdx.x != 0 || blockIdx.x != 0) return;
  const float* We[3] = {We0, We1, We2};
  const float* ae[3] = {ae0, ae1, ae2};
  float m0 = easum[0] * invE, m1 = easum[1] * invE, m2 = easum[2] * invE;
  for (int i = 0; i < 3; ++i) {
    float w[3];
    for (int j = 0; j < 3; ++j) {
      float s = 0.f;
      for (int c = 0; c < 128; ++c) s += We[i][j * 128 + c] * ae[i][c];
      w[j] = s;
      wattOut[i * 4 + j] = s;
    }
    wattOut[i * 4 + 3] = m0 * w[0] + m1 * w[1] + m2 * w[2];  // self-loop a_e
  }
}

// ---------- a_src / a_dst = <xh[n], att_{src,dst}> ; one wave per node ----------
__global__ void node_atts(const float* __restrict__ xh,
                          const float* __restrict__ att_src, const float* __restrict__ att_dst,
                          float* __restrict__ asrc, float* __restrict__ adst, int N) {
  int wv = blockIdx.x * (blockDim.x >> 5) + (threadIdx.x >> 5);
  int lane = threadIdx.x & 31;
  if (wv >= N) return;
  float4 a  = ((const float4*)(xh + (size_t)wv * 128))[lane];
  float4 s4 = ((const float4*)att_src)[lane];
  float4 d4 = ((const float4*)att_dst)[lane];
  float ps = a.x * s4.x + a.y * s4.y + a.z * s4.z + a.w * s4.w;
  float pd = a.x * d4.x + a.y * d4.y + a.z * d4.z + a.w * d4.w;
  ps = wave_sum(ps);
  pd = wave_sum(pd);
  if (lane == 0) { asrc[wv] = ps; adst[wv] = pd; }
}

// ---------- alpha = leaky_relu(a_src[s]+a_dst[d]+a_e) ; segment max ----------
__global__ void edge_alpha(const int* __restrict__ src, const int* __restrict__ dst,
                           const float* __restrict__ ea, const float* __restrict__ asrc,
                           const float* __restrict__ adst, const float* __restrict__ watt,
                           float* __restrict__ alpha, unsigned* __restrict__ amax,
                           int E, int N) {
  int e = blockIdx.x * blockDim.x + threadIdx.x;
  if (e >= E + N) return;
  int s, d; float aeterm;
  if (e < E) {
    s = src[e]; d = dst[e];
    aeterm = ea[(size_t)e * 3 + 0] * watt[0] + ea[(size_t)e * 3 + 1] * watt[1] +
             ea[(size_t)e * 3 + 2] * watt[2];
  } else {
    s = d = e - E;
    aeterm = watt[3];
  }
  float x = asrc[s] + adst[d] + aeterm;
  x = (x > 0.f) ? x : NEG_SLOPE * x;
  alpha[e] = x;
  atomicMax(&amax[d], ord_f32(x));
}

// ---------- ex = exp(alpha - amax[d]) ; segment sum ----------
__global__ void edge_ex(const int* __restrict__ dst, float* __restrict__ alpha,
                        const unsigned* __restrict__ amax, float* __restrict__ den,
                        int E, int N) {
  int e = blockIdx.x * blockDim.x + threadIdx.x;
  if (e >= E + N) return;
  int d = (e < E) ? dst[e] : (e - E);
  float v = expf(alpha[e] - deord_f32(amax[d]));
  alpha[e] = v;
  atomicAdd(&den[d], v);
}

// ---------- out[d] += xh[s] * (ex / den[d]) ; one wave per edge ----------
__global__ void edge_agg(const int* __restrict__ src, const int* __restrict__ dst,
                         const float* __restrict__ ex, const float* __restrict__ den,
                         const float* __restrict__ xh, float* __restrict__ agg,
                         int E, int N) {
  int wv = blockIdx.x * (blockDim.x >> 5) + (threadIdx.x >> 5);
  int lane = threadIdx.x & 31;
  if (wv >= E + N) return;
  int s, d;
  if (wv < E) { s = src[wv]; d = dst[wv]; } else { s = d = wv - E; }
  float w = ex[wv] / den[d];
  const float* xs = xh + (size_t)s * 128;
  float* od = agg + (size_t)d * 128;
#pragma unroll
  for (int c = 0; c < 4; ++c) {
    int idx = lane + c * 32;
    atomicAdd(&od[idx], xs[idx] * w);
  }
}

// ---------- agg -> h : h = maybe_elu(gamma*(agg+b)*invs + beta + resid) in place ----------
__global__ void gat_post(float* __restrict__ agg, const float* __restrict__ bgat,
                         const float* __restrict__ gamma, const float* __restrict__ beta,
                         const float* __restrict__ resid, long total, int do_elu) {
  long i = (long)blockIdx.x * blockDim.x + threadIdx.x;
  if (i >= total) return;
  int c = (int)(i & 127);
  float v = agg[i] + bgat[c];
  v = gamma[c] * v * BN_INVS + beta[c] + resid[i];
  if (do_elu) v = (v > 0.f) ? v : (expf(v) - 1.f);
  agg[i] = v;
}

// ---------- scalar heads: out[n] = f(<in[n], W>(K) + b) ; one wave per node ----------
// mode: 0 sigmoid, 1 softplus, 2 tanh
__global__ void head_out(const float* __restrict__ in, const float* __restrict__ W,
                         const float* __restrict__ b, float* __restrict__ out,
                         int N, int K, int mode) {
  int wv = blockIdx.x * (blockDim.x >> 5) + (threadIdx.x >> 5);
  int lane = threadIdx.x & 31;
  if (wv >= N) return;
  float p = 0.f;
  for (int c = lane; c < K; c += 32) p += in[(size_t)wv * K + c] * W[c];
  p = wave_sum(p);
  if (lane == 0) {
    float x = p + b[0];
    float y;
    if (mode == 0)      y = 1.f / (1.f + expf(-x));
    else if (mode == 1) y = (x > 20.f) ? x : log1pf(expf(x));
    else                y = tanhf(x);
    out[wv] = y;
  }
}

// ===================== host launch =====================
extern "C" void kernel_launch(void* const* d_in, const int* in_sizes, int n_in,
                              void* d_out, int out_size, void* d_ws, size_t ws_size,
                              hipStream_t stream) {
  const float* x    = (const float*)d_in[0];
  const int*   eidx = (const int*)d_in[1];
  const float* ea   = (const float*)d_in[2];

  const int N = in_sizes[0] / 64;      // F_NODE = 64
  const int E = in_sizes[1] / 2;
  const int Etot = E + N;
  const int* src = eidx;
  const int* dst = eidx + E;

  // params (flattened setup_inputs insertion order)
  const float* enc_W1 = (const float*)d_in[3];
  const float* enc_b1 = (const float*)d_in[4];
  const float* enc_W2 = (const float*)d_in[5];
  const float* enc_b2 = (const float*)d_in[6];
  const float *gatW[3], *gatb[3], *attS[3], *attD[3], *We[3], *attE[3];
  for (int i = 0; i < 3; ++i) {
    int b = 7 + 6 * i;
    gatW[i] = (const float*)d_in[b + 0];
    gatb[i] = (const float*)d_in[b + 1];
    attS[i] = (const float*)d_in[b + 2];
    attD[i] = (const float*)d_in[b + 3];
    We[i]   = (const float*)d_in[b + 4];
    attE[i] = (const float*)d_in[b + 5];
  }
  const float *gamma[3], *beta[3], *skW[3], *skb[3];
  for (int i = 0; i < 3; ++i) {
    gamma[i] = (const float*)d_in[25 + 2 * i];
    beta[i]  = (const float*)d_in[26 + 2 * i];
    skW[i]   = (const float*)d_in[31 + 2 * i];
    skb[i]   = (const float*)d_in[32 + 2 * i];
  }
  const float* risk_W1 = (const float*)d_in[37];
  const float* risk_b1 = (const float*)d_in[38];
  const float* risk_W2 = (const float*)d_in[39];
  const float* risk_b2 = (const float*)d_in[40];
  const float* risk_W3 = (const float*)d_in[41];
  const float* risk_b3 = (const float*)d_in[42];
  const float* vol_W1  = (const float*)d_in[43];
  const float* vol_b1  = (const float*)d_in[44];
  const float* vol_W2  = (const float*)d_in[45];
  const float* vol_b2  = (const float*)d_in[46];
  const float* ret_W1  = (const float*)d_in[47];
  const float* ret_b1  = (const float*)d_in[48];
  const float* ret_W2  = (const float*)d_in[49];
  const float* ret_b2  = (const float*)d_in[50];

  // workspace carve (floats)
  float* ws       = (float*)d_ws;
  float* easum    = ws;                       // 4
  float* watt     = ws + 4;                   // 12
  float* asrc     = ws + 16;                  // N
  float* adst     = asrc + N;                 // N
  unsigned* amax  = (unsigned*)(adst + N);    // N
  float* den      = (float*)amax + N;         // N
  float* alpha    = den + N;                  // Etot (alpha, then ex in place)
  float* big      = alpha + Etot;
  const size_t HN = (size_t)N * 128;
  float* hb0  = big;            // encoder output
  float* hb1  = hb0 + HN;       // layer-0 output
  float* hb2  = hb1 + HN;       // layer-1 output
  float* bufX = hb2 + HN;       // xh / head temp
  float* bufR = bufX + HN;      // residual / head temp

  float* out_risk = (float*)d_out;
  float* out_vol  = out_risk + N;
  float* out_ret  = out_vol + N;
  float* out_h    = out_ret + N;              // final h lives in d_out

  auto gemm = [&](const float* A, const float* W, const float* bias, float* C,
                  int M, int K, int Nout, int act) {
    int smt = ((M / 16) + 3) / 4;             // 64-row supertiles
    int snt = Nout / 32;                      // 32-col supertiles
    int blocks = (smt * snt + 7) / 8;
    if (K == 64 && Nout == 128)
      gemm_f32_wmma<64, 128><<<blocks, 256, 0, stream>>>(A, W, bias, C, M, act);
    else if (K == 128 && Nout == 128)
      gemm_f32_wmma<128, 128><<<blocks, 256, 0, stream>>>(A, W, bias, C, M, act);
    else if (K == 128 && Nout == 64)
      gemm_f32_wmma<128, 64><<<blocks, 256, 0, stream>>>(A, W, bias, C, M, act);
    else if (K == 64 && Nout == 32)
      gemm_f32_wmma<64, 32><<<blocks, 256, 0, stream>>>(A, W, bias, C, M, act);
    else if (K == 128 && Nout == 32)
      gemm_f32_wmma<128, 32><<<blocks, 256, 0, stream>>>(A, W, bias, C, M, act);
  };
  const int TB = 256;
  int nblk  = (N + TB - 1) / TB;
  int eblk  = (Etot + TB - 1) / TB;
  int nwblk = (N + 7) / 8;        // wave-per-node
  int ewblk = (Etot + 7) / 8;     // wave-per-edge
  long hn = (long)HN;
  int hblk = (int)((hn + TB - 1) / TB);

  // edge-attr mean + per-layer attention projections
  fill_f32<<<1, 16, 0, stream>>>(easum, 0.f, 4);
  ea_reduce<<<256, 256, 0, stream>>>(ea, easum, E);
  prep_watt<<<1, 32, 0, stream>>>(easum, 1.f / (float)E,
                                  We[0], attE[0], We[1], attE[1], We[2], attE[2], watt);

  // encoder: h0 = relu(x@W1+b1)@W2 + b2
  gemm(x, enc_W1, enc_b1, bufX, N, 64, 128, 1);
  gemm(bufX, enc_W2, enc_b2, hb0, N, 128, 128, 0);

  float* h  = hb0;
  float* hp = hb0;
  float* htgt[3] = { hb1, hb2, out_h };

  for (int i = 0; i < 3; ++i) {
    const float* rsrc = (i == 0) ? h : hp;
    gemm(rsrc, skW[i], skb[i], bufR, N, 128, 128, 0);        // residual
    gemm(h, gatW[i], nullptr, bufX, N, 128, 128, 0);         // xh = h @ W
    node_atts<<<nwblk, 256, 0, stream>>>(bufX, attS[i], attD[i], asrc, adst, N);
    fill_u32<<<nblk, TB, 0, stream>>>(amax, 0u, N);
    fill_f32<<<nblk, TB, 0, stream>>>(den, 0.f, N);
    fill_f32<<<hblk, TB, 0, stream>>>(htgt[i], 0.f, hn);
    edge_alpha<<<eblk, TB, 0, stream>>>(src, dst, ea, asrc, adst, watt + 4 * i,
                                        alpha, amax, E, N);
    edge_ex<<<eblk, TB, 0, stream>>>(dst, alpha, amax, den, E, N);
    edge_agg<<<ewblk, 256, 0, stream>>>(src, dst, alpha, den, bufX, htgt[i], E, N);
    gat_post<<<hblk, TB, 0, stream>>>(htgt[i], gatb[i], gamma[i], beta[i], bufR,
                                      hn, (i < 2) ? 1 : 0);
    hp = h;
    h = htgt[i];
  }

  // heads (h == out_h)
  gemm(h, risk_W1, risk_b1, bufX, N, 128, 64, 1);
  gemm(bufX, risk_W2, risk_b2, bufR, N, 64, 32, 1);
  head_out<<<nwblk, 256, 0, stream>>>(bufR, risk_W3, risk_b3, out_risk, N, 32, 0);

  gemm(h, vol_W1, vol_b1, bufR, N, 128, 32, 1);
  head_out<<<nwblk, 256, 0, stream>>>(bufR, vol_W2, vol_b2, out_vol, N, 32, 1);

  gemm(h, ret_W1, ret_b1, bufR, N, 128, 32, 1);
  head_out<<<nwblk, 256, 0, stream>>>(bufR, ret_W2, ret_b2, out_ret, N, 32, 2);
}